// BQNN_simulation_29686813950725
// MI455X (gfx1250) — compile-verified
//
#include <hip/hip_runtime.h>
#include <math.h>

#define BATCH 32768
#define BLOCK 256

__global__ __launch_bounds__(BLOCK) void bqnn_kernel(
    const float* __restrict__ x,            // (32768, 12)
    const float* __restrict__ params,       // (30,)  [0:15]=phi, [15:30]=theta
    const float* __restrict__ output_phase, // (6,)
    const float* __restrict__ param_phi,    // (4,)
    const float* __restrict__ param_theta,  // (4,)
    const float* __restrict__ input_k,      // (12,)
    const float* __restrict__ input_b,      // (12,)
    float* __restrict__ out)                // (32768, 20)
{
    __shared__ float  Sr[6][3];
    __shared__ float  Si[6][3];
    __shared__ float  PC[4][6];             // param-MZI coeffs: ar ai br bi st ct
    __shared__ float4 lds_x[BLOCK * 3];     // staged input tile (12 KB)

    const int i = blockIdx.x * BLOCK + threadIdx.x;

    // ---- kick off async global->LDS staging of this block's x tile -------
    // imm offset is added to both the LDS and the global address  (same imm
    // offset advances both sides), so one (lds,addr) pair covers 48 B.
    {
        unsigned lds_off =
            (unsigned)(unsigned long long)(uintptr_t)&lds_x[threadIdx.x * 3];
        unsigned long long gaddr =
            (unsigned long long)(uintptr_t)x + (unsigned long long)i * 48ull;
        asm volatile(
            "global_load_async_to_lds_b128 %0, %1, off\n\t"
            "global_load_async_to_lds_b128 %0, %1, off offset:16\n\t"
            "global_load_async_to_lds_b128 %0, %1, off offset:32"
            :
            : "v"(lds_off), "v"(gaddr)
            : "memory");
    }

    // ---- per-block constant setup, overlapped with the async transfer ----
    if (threadIdx.x == 0) {
        const int cm[15] = {0,2,4,1,3,0,2,4,1,3,0,2,4,1,3};
        const int cn[15] = {1,3,5,2,4,1,3,5,2,4,1,3,5,2,4};
        float Tr[6][3], Ti[6][3];
        #pragma unroll
        for (int r = 0; r < 6; ++r)
            #pragma unroll
            for (int c = 0; c < 3; ++c) { Tr[r][c] = (r == c) ? 1.f : 0.f; Ti[r][c] = 0.f; }
        #pragma unroll
        for (int k = 0; k < 15; ++k) {
            float st, ct, sp, cp;
            __sincosf(params[15 + k], &st, &ct);   // theta
            __sincosf(params[k],      &sp, &cp);   // e^{i phi} = cp + i sp
            const float ar = cp * ct, ai = sp * ct;
            const float br = cp * st, bi = sp * st;
            const int m = cm[k], n = cn[k];
            #pragma unroll
            for (int c = 0; c < 3; ++c) {
                const float umr = Tr[m][c], umi = Ti[m][c];
                const float unr = Tr[n][c], uni = Ti[n][c];
                Tr[m][c] = ar * umr - ai * umi - st * unr;
                Ti[m][c] = ai * umr + ar * umi - st * uni;
                Tr[n][c] = br * umr - bi * umi + ct * unr;
                Ti[n][c] = bi * umr + br * umi + ct * uni;
            }
        }
        #pragma unroll
        for (int r = 0; r < 6; ++r) {
            float so, co; __sincosf(output_phase[r], &so, &co);
            #pragma unroll
            for (int c = 0; c < 3; ++c) {
                Sr[r][c] = co * Tr[r][c] - so * Ti[r][c];
                Si[r][c] = so * Tr[r][c] + co * Ti[r][c];
            }
        }
        // batch-constant ansatz-MZI coefficients (k = 3,4,8,9 -> p = 0..3)
        #pragma unroll
        for (int p = 0; p < 4; ++p) {
            float st, ct, sp, cp;
            __sincosf(param_theta[p], &st, &ct);
            __sincosf(param_phi[p],   &sp, &cp);
            PC[p][0] = cp * ct; PC[p][1] = sp * ct;
            PC[p][2] = cp * st; PC[p][3] = sp * st;
            PC[p][4] = st;      PC[p][5] = ct;
        }
    }

    // ---- wait for async tile + setup, then proceed ------------------------
    asm volatile("s_wait_asynccnt 0" ::: "memory");
    __syncthreads();

    const float4 a0 = lds_x[threadIdx.x * 3 + 0];
    const float4 a1 = lds_x[threadIdx.x * 3 + 1];
    const float4 a2 = lds_x[threadIdx.x * 3 + 2];
    float xs[12] = {a0.x, a0.y, a0.z, a0.w, a1.x, a1.y, a1.z, a1.w,
                    a2.x, a2.y, a2.z, a2.w};
    #pragma unroll
    for (int j = 0; j < 12; ++j) xs[j] = fmaf(xs[j], input_k[j], input_b[j]);

    // ---- evolve the 6x3 constant matrix through the 10-MZI ansatz ---------
    float Ur[6][3], Ui[6][3];
    #pragma unroll
    for (int r = 0; r < 6; ++r)
        #pragma unroll
        for (int c = 0; c < 3; ++c) { Ur[r][c] = Sr[r][c]; Ui[r][c] = Si[r][c]; }

    const int am[10]   = {0,2,4,1,3,0,2,4,1,3};
    const int an[10]   = {1,3,5,2,4,1,3,5,2,4};
    const int tIdx[10] = {3,4,5,-1,-1,9,10,11,-1,-1};   // xs index for theta
    const int pIdx[10] = {0,1,2,-1,-1,6,7,8,-1,-1};     // xs index for phi
    const int pPar[10] = {-1,-1,-1,0,1,-1,-1,-1,2,3};   // PC row for param MZIs

    #pragma unroll
    for (int k = 0; k < 10; ++k) {
        float ar, ai, br, bi, st, ct;
        if (pPar[k] >= 0) {
            const int p = pPar[k];
            ar = PC[p][0]; ai = PC[p][1]; br = PC[p][2];
            bi = PC[p][3]; st = PC[p][4]; ct = PC[p][5];
        } else {
            float sp, cp;
            __sincosf(xs[tIdx[k]], &st, &ct);
            __sincosf(xs[pIdx[k]], &sp, &cp);
            ar = cp * ct; ai = sp * ct;
            br = cp * st; bi = sp * st;
        }
        const int m = am[k], n = an[k];
        #pragma unroll
        for (int c = 0; c < 3; ++c) {
            const float umr = Ur[m][c], umi = Ui[m][c];
            const float unr = Ur[n][c], uni = Ui[n][c];
            Ur[m][c] = ar * umr - ai * umi - st * unr;
            Ui[m][c] = ai * umr + ar * umi - st * uni;
            Ur[n][c] = br * umr - bi * umi + ct * unr;
            Ui[n][c] = bi * umr + br * umi + ct * uni;
        }
    }

    // ---- 20 permanents of 3x3 complex submatrices -------------------------
    const int RA[20] = {0,0,0,0,0,0,0,0,0,0,1,1,1,1,1,1,2,2,2,3};
    const int RB[20] = {1,1,1,1,2,2,2,3,3,4,2,2,2,3,3,4,3,3,4,4};
    const int RC[20] = {2,3,4,5,3,4,5,4,5,5,3,4,5,4,5,5,4,5,5,5};

    float m2[20];
    float sum = 0.f;
    #pragma unroll
    for (int o = 0; o < 20; ++o) {
        const int ra = RA[o], rb = RB[o], rc = RC[o];
        const float u0r = Ur[ra][0], u0i = Ui[ra][0];
        const float u1r = Ur[ra][1], u1i = Ui[ra][1];
        const float u2r = Ur[ra][2], u2i = Ui[ra][2];
        const float v0r = Ur[rb][0], v0i = Ui[rb][0];
        const float v1r = Ur[rb][1], v1i = Ui[rb][1];
        const float v2r = Ur[rb][2], v2i = Ui[rb][2];
        const float w0r = Ur[rc][0], w0i = Ui[rc][0];
        const float w1r = Ur[rc][1], w1i = Ui[rc][1];
        const float w2r = Ur[rc][2], w2i = Ui[rc][2];
        const float d0r = v1r*w2r - v1i*w2i + v2r*w1r - v2i*w1i;
        const float d0i = v1r*w2i + v1i*w2r + v2r*w1i + v2i*w1r;
        const float d1r = v0r*w2r - v0i*w2i + v2r*w0r - v2i*w0i;
        const float d1i = v0r*w2i + v0i*w2r + v2r*w0i + v2i*w0r;
        const float d2r = v0r*w1r - v0i*w1i + v1r*w0r - v1i*w0i;
        const float d2i = v0r*w1i + v0i*w1r + v1r*w0i + v1i*w0r;
        const float pr = u0r*d0r - u0i*d0i + u1r*d1r - u1i*d1i + u2r*d2r - u2i*d2i;
        const float pi = u0r*d0i + u0i*d0r + u1r*d1i + u1i*d1r + u2r*d2i + u2i*d2r;
        const float mm = pr*pr + pi*pi;
        m2[o] = mm;
        sum += mm;
    }

    const float inv = 1.0f / fmaxf(sqrtf(sum), 1e-12f);
    float res[20];
    #pragma unroll
    for (int o = 0; o < 20; ++o) res[o] = sqrtf(m2[o]) * inv;

    float4* op = reinterpret_cast<float4*>(out + (size_t)i * 20);
    op[0] = make_float4(res[0],  res[1],  res[2],  res[3]);
    op[1] = make_float4(res[4],  res[5],  res[6],  res[7]);
    op[2] = make_float4(res[8],  res[9],  res[10], res[11]);
    op[3] = make_float4(res[12], res[13], res[14], res[15]);
    op[4] = make_float4(res[16], res[17], res[18], res[19]);
}

extern "C" void kernel_launch(void* const* d_in, const int* in_sizes, int n_in,
                              void* d_out, int out_size, void* d_ws, size_t ws_size,
                              hipStream_t stream) {
    const float* x            = (const float*)d_in[0];
    const float* params       = (const float*)d_in[1];
    const float* output_phase = (const float*)d_in[2];
    const float* param_phi    = (const float*)d_in[3];
    const float* param_theta  = (const float*)d_in[4];
    const float* input_k      = (const float*)d_in[5];
    const float* input_b      = (const float*)d_in[6];
    float* out = (float*)d_out;
    (void)in_sizes; (void)n_in; (void)out_size; (void)d_ws; (void)ws_size;

    dim3 grid(BATCH / BLOCK), block(BLOCK);
    bqnn_kernel<<<grid, block, 0, stream>>>(x, params, output_phase, param_phi,
                                            param_theta, input_k, input_b, out);
}